// SymmetricLightweightConv_74904229643067
// MI455X (gfx1250) — compile-verified
//
#include <hip/hip_runtime.h>
#include <stdint.h>

// SymmetricLightweightConv: out[b,c,t] = sum_k w[f(c),k] * x[b,c,t+k-1]
//   f(c) = c (c<256) else 511-c;  w = softmax(weight,axis=-1)*drop_mask
// HBM-bandwidth-bound streaming kernel (AI ~0.75 FLOP/B, ~11.5us floor at
// 23.3 TB/s). One 256-thread block per (b,c) row; row staged to LDS via
// CDNA5 async global->LDS B128 copies (ASYNCcnt path), NT hints both ways.

#define T_LEN 4096
#define C_TOT 512
#define HALF  256

typedef float f32x4 __attribute__((ext_vector_type(4)));

__global__ __launch_bounds__(256) void symconv_k(
    const float* __restrict__ x,
    const float* __restrict__ wgt,
    const float* __restrict__ dmask,
    float* __restrict__ y)
{
    // 4-float front pad keeps async B128 destination 16B-aligned; slots sr[-1]
    // and sr[T_LEN] hold the zero halo for 'same' padding.
    __shared__ float s[4 + T_LEN + 4];
    float* sr = s + 4;

    const int row = blockIdx.x;            // 0 .. B*C-1, layout [B,C,T]
    const int c   = row & (C_TOT - 1);
    const int f   = (c < HALF) ? c : (C_TOT - 1 - c);

    const float* xrow = x + (size_t)row * T_LEN;
    float*       yrow = y + (size_t)row * T_LEN;

    if (threadIdx.x == 0) { sr[-1] = 0.0f; sr[T_LEN] = 0.0f; }

    // Issue the async DMA first so the softmax below hides under it.
    // Async-stage the 16KB row into LDS: 4 rounds x 256 lanes x 16B.
    // Generic LDS pointer low 32 bits == LDS byte offset (ISA flat->LDS rule).
    // Input lines are touched exactly once -> NT so they don't thrash L2.
    const uint32_t lds0 = (uint32_t)(uintptr_t)sr;
#pragma unroll
    for (int r = 0; r < 4; ++r) {
        const uint32_t fo = ((uint32_t)r * 256u + threadIdx.x) * 4u; // float idx
        const uint32_t la = lds0 + fo * 4u;                          // LDS bytes
        const float*   ga = xrow + fo;                               // global
        asm volatile("global_load_async_to_lds_b128 %0, %1, off th:TH_LOAD_NT"
                     :: "v"(la), "v"(ga)
                     : "memory");
    }

    // Per-channel 3-tap filter: softmax over K, then inverted-dropout mask.
    // f is uniform per block -> these go through the scalar (s_load) path.
    const float a0 = wgt[3 * f + 0];
    const float a1 = wgt[3 * f + 1];
    const float a2 = wgt[3 * f + 2];
    const float mx = fmaxf(a0, fmaxf(a1, a2));
    const float e0 = __expf(a0 - mx);
    const float e1 = __expf(a1 - mx);
    const float e2 = __expf(a2 - mx);
    const float inv = 1.0f / (e0 + e1 + e2);
    const float w0 = e0 * inv * dmask[3 * f + 0];
    const float w1 = e1 * inv * dmask[3 * f + 1];
    const float w2 = e2 * inv * dmask[3 * f + 2];

    // Each wave drains its own ASYNCcnt before the barrier -> LDS data is
    // visible workgroup-wide after __syncthreads().
    asm volatile("s_wait_asynccnt 0" ::: "memory");
    __syncthreads();

    // 16 outputs per thread; 18-wide input window from LDS (halo pre-zeroed).
    const int t0 = (int)threadIdx.x * 16;
    float in[18];
    in[0] = sr[t0 - 1];
#pragma unroll
    for (int v = 0; v < 4; ++v) {
        const f32x4 q = *(const f32x4*)(sr + t0 + 4 * v);   // ds_load_b128
        in[1 + 4 * v + 0] = q.x;
        in[1 + 4 * v + 1] = q.y;
        in[1 + 4 * v + 2] = q.z;
        in[1 + 4 * v + 3] = q.w;
    }
    in[17] = sr[t0 + 16];

#pragma unroll
    for (int v = 0; v < 4; ++v) {
        f32x4 o;
        o.x = fmaf(w0, in[4 * v + 0], fmaf(w1, in[4 * v + 1], w2 * in[4 * v + 2]));
        o.y = fmaf(w0, in[4 * v + 1], fmaf(w1, in[4 * v + 2], w2 * in[4 * v + 3]));
        o.z = fmaf(w0, in[4 * v + 2], fmaf(w1, in[4 * v + 3], w2 * in[4 * v + 4]));
        o.w = fmaf(w0, in[4 * v + 3], fmaf(w1, in[4 * v + 4], w2 * in[4 * v + 5]));
        // Pure streaming output: nontemporal B128 store keeps L2 for inputs.
        __builtin_nontemporal_store(o, (f32x4*)(yrow + t0 + 4 * v));
    }
}

extern "C" void kernel_launch(void* const* d_in, const int* in_sizes, int n_in,
                              void* d_out, int out_size, void* d_ws, size_t ws_size,
                              hipStream_t stream) {
    const float* x     = (const float*)d_in[0];  // [16, 512, 4096] f32
    const float* wgt   = (const float*)d_in[1];  // [256, 1, 3] f32
    const float* dmask = (const float*)d_in[2];  // [256, 1, 3] f32
    float* y = (float*)d_out;                    // [16, 512, 4096] f32

    const int rows = in_sizes[0] / T_LEN;        // B*C = 8192
    hipLaunchKernelGGL(symconv_k, dim3(rows), dim3(256), 0, stream,
                       x, wgt, dmask, y);
}